// newNormal_80968723464204
// MI455X (gfx1250) — compile-verified
//
#include <hip/hip_runtime.h>

typedef __attribute__((ext_vector_type(2))) float v2f;
typedef __attribute__((ext_vector_type(4))) float f4;
typedef __attribute__((ext_vector_type(8))) float v8f;

#define EPSV 1e-8f
#define DIVC 512.0f   // hardcoded DIV_CONST from the reference

// Fixed problem shape from the reference: x is [B=16, C=512, T=8192] f32.
// Compile-time constants so channel-row strides become 24-bit IOFFSET
// immediates on the global loads (4*T*4B = 128KB per WMMA step, max 928KB
// within one unrolled group -- well inside the +-8MB immediate range).
constexpr int kT = 8192;
constexpr int kC = 512;

// ---------------------------------------------------------------------------
// Kernel 1: per-(b,t) channel sums of x and x*x using V_WMMA_F32_16X16X4_F32.
// A = all-ones 16x4  =>  D[m,n] += sum_k B[k,n]  (exact f32 accumulation).
// Each wave owns 16 t-columns; lanes 0-15 carry K={0,1} rows of B in the two
// fragment VGPRs, lanes 16-31 carry K={2,3}. With A==1 the result is the sum
// over all 4 channels per step regardless of K-slot order.
// Uniform SGPR base + unsigned 32-bit lane offset -> SADDR-form loads with
// immediate offsets; one scalar-width offset add per 32 channels.
// ---------------------------------------------------------------------------
__global__ __launch_bounds__(256) void colsum_wmma_kernel(
    const float* __restrict__ x,
    float* __restrict__ sum_out,
    float* __restrict__ sq_out)
{
  const int lane = threadIdx.x & 31;
  const int wave = threadIdx.x >> 5;
  const int n    = lane & 15;        // column (t) within the 16-wide tile
  const int half = lane >> 4;        // 0: K rows 0..1, 1: K rows 2..3
  const int b    = blockIdx.y;
  const int t0   = (blockIdx.x * 8 + wave) << 4;   // 16 t's per wave

  const float* xb = x + (size_t)b * kC * kT;       // uniform across block
  unsigned off = (unsigned)((half << 1) * kT + t0 + n);  // per-lane, 32-bit

  v8f acc  = {};   // D/C accumulator for sum(x)
  v8f acc2 = {};   // D/C accumulator for sum(x*x)
  const v2f ones = {1.0f, 1.0f};

  for (int g = 0; g < kC / 32; ++g) {  // 16 groups x 8 WMMA-steps (32 channels)
#pragma unroll
    for (int u = 0; u < 8; ++u) {
      const float e0 = xb[off + (unsigned)(4 * u * kT)];       // ch 4u+2*half
      const float e1 = xb[off + (unsigned)((4 * u + 1) * kT)]; // ch 4u+2*half+1
      const v2f bv = {e0, e1};
      const v2f bs = {e0 * e0, e1 * e1};
      // (neg_a, A, neg_b, B, c_mod, C, reuse_a, reuse_b)
      acc  = __builtin_amdgcn_wmma_f32_16x16x4_f32(false, ones, false, bv,
                                                   (short)0, acc,  false, false);
      acc2 = __builtin_amdgcn_wmma_f32_16x16x4_f32(false, ones, false, bs,
                                                   (short)0, acc2, false, false);
    }
    off += 32u * kT;
  }

  // Every D row holds the same column sums; take row M=0 (VGPR0, lanes 0-15).
  if (lane < 16) {
    const size_t o = (size_t)b * kT + (size_t)(t0 + n);
    sum_out[o] = acc[0];
    sq_out[o]  = acc2[0];
  }
}

// ---------------------------------------------------------------------------
// Kernel 2: per-batch inclusive scan over T of (chansum, chansumsq),
// converted in place to mean[b,t] and rstd[b,t]. One block per batch,
// 256 threads x 32 contiguous elements. Block-level combine uses a
// Hillis-Steele log-step LDS scan (8 steps); combination order is fixed ->
// deterministic output every call.
// ---------------------------------------------------------------------------
__global__ __launch_bounds__(256) void scan_kernel(
    float* __restrict__ sum_buf,   // in: channel sums  -> out: mean
    float* __restrict__ sq_buf)    // in: channel sqsum -> out: rstd
{
  const int b   = blockIdx.x;
  const int tid = threadIdx.x;
  constexpr int CHUNK = kT / 256;  // 32
  __shared__ float s1[256];
  __shared__ float s2[256];
  const size_t base = (size_t)b * kT + (size_t)tid * CHUNK;

  // Phase 1: per-thread chunk totals
  float tot1 = 0.0f, tot2 = 0.0f;
  for (int i = 0; i < CHUNK; ++i) {
    tot1 += sum_buf[base + i];
    tot2 += sq_buf[base + i];
  }
  s1[tid] = tot1;
  s2[tid] = tot2;
  __syncthreads();

  // Phase 2: inclusive Hillis-Steele scan of the 256 partials
#pragma unroll
  for (int d = 1; d < 256; d <<= 1) {
    const float a1 = (tid >= d) ? s1[tid - d] : 0.0f;
    const float a2 = (tid >= d) ? s2[tid - d] : 0.0f;
    __syncthreads();
    s1[tid] += a1;
    s2[tid] += a2;
    __syncthreads();
  }

  // Exclusive prefix for this thread = inclusive prefix of tid-1
  const float off1 = (tid > 0) ? s1[tid - 1] : 0.0f;
  const float off2 = (tid > 0) ? s2[tid - 1] : 0.0f;

  // Phase 3: local running prefix -> mean / rstd (in place)
  float run1 = off1;
  float run2 = off2;
  for (int i = 0; i < CHUNK; ++i) {
    run1 += sum_buf[base + i];
    run2 += sq_buf[base + i];
    const float inv  = 1.0f / (DIVC * (float)(tid * CHUNK + i + 1));
    const float mean = run1 * inv;
    const float msq  = run2 * inv;
    const float var  = msq - mean * mean;
    const float rstd = rsqrtf(var + EPSV);
    sum_buf[base + i] = mean;
    sq_buf[base + i]  = rstd;
  }
}

// ---------------------------------------------------------------------------
// Kernel 3: out = (x - mean[b,t]) * rstd[b,t] * gains[c] + bias[c]
// grid = (T/1024, C, B): no div/mod; float4 (b128) traffic; nontemporal
// hints on the 256MB x-read and 256MB out-write streams so they don't evict
// the C-times-reused mean/rstd from L2; prefetch the read stream.
// ---------------------------------------------------------------------------
__global__ __launch_bounds__(256) void norm_kernel(
    const float* __restrict__ x,
    const float* __restrict__ gains,
    const float* __restrict__ bias,
    const float* __restrict__ mean,
    const float* __restrict__ rstd,
    float* __restrict__ out)
{
  const int t = (int)((blockIdx.x * blockDim.x + threadIdx.x) << 2);
  const int c = blockIdx.y;
  const int b = blockIdx.z;
  const size_t xo = ((size_t)b * kC + c) * kT + t;
  const size_t mo = (size_t)b * kT + t;

  // CDNA5 global_prefetch_b8: pull the read stream ahead
  __builtin_prefetch(x + xo + 4096, 0, 0);

  const f4 xv = __builtin_nontemporal_load((const f4*)(x + xo));
  const f4 m  = *(const f4*)(mean + mo);   // reused C times: keep cached
  const f4 r  = *(const f4*)(rstd + mo);
  const float g  = gains[c];
  const float bi = bias[c];

  f4 o = (xv - m) * r * g + bi;
  __builtin_nontemporal_store(o, (f4*)(out + xo));
}

// ---------------------------------------------------------------------------
extern "C" void kernel_launch(void* const* d_in, const int* in_sizes, int n_in,
                              void* d_out, int out_size, void* d_ws, size_t ws_size,
                              hipStream_t stream)
{
  (void)n_in; (void)out_size; (void)ws_size;

  const float* x     = (const float*)d_in[0];
  const float* gains = (const float*)d_in[1];
  const float* bias  = (const float*)d_in[2];
  float* out = (float*)d_out;

  const int B = in_sizes[0] / (kC * kT);   // 16

  float* sum_buf = (float*)d_ws;                    // B*T floats -> mean
  float* sq_buf  = sum_buf + (size_t)B * kT;        // B*T floats -> rstd

  // Pass 1: WMMA channel reduction. 8 waves/block, 16 t per wave -> 128 t/block
  dim3 g1(kT / 128, B, 1);
  colsum_wmma_kernel<<<g1, 256, 0, stream>>>(x, sum_buf, sq_buf);

  // Pass 2: per-batch scan -> mean/rstd (in place)
  scan_kernel<<<B, 256, 0, stream>>>(sum_buf, sq_buf);

  // Pass 3: streaming normalize
  dim3 g3(kT / 1024, kC, B);
  norm_kernel<<<g3, 256, 0, stream>>>(x, gains, bias, sum_buf, sq_buf, out);
}